// DVAE_81286551044786
// MI455X (gfx1250) — compile-verified
//
#include <hip/hip_runtime.h>

// ---------------------------------------------------------------------------
// Geometry constants from the reference
// ---------------------------------------------------------------------------
#define GH 512
#define GW 512
#define DH 256
#define DW 256
#define BATCH 64
#define NCH 2
#define PLANE (DH * DW)                 // 65536
#define PLANES (BATCH * NCH)            // 128
#define FIELD_ELEMS (PLANES * PLANE)    // 8388608 floats per field buffer
#define RAD 9                           // int(3*sigma), sigma = 3
#define TAPS 19                         // 2*RAD+1
#define KCH 9                           // ceil((16 + 2*RAD) / 4) WMMA K-chunks
#define TPW 4                           // tiles processed per wave (blur)
#define DT_C (1.0f / 7.0f)              // 1/(NTP-1)
#define G_SP (511.0f / 255.0f)          // dgs index -> full-grid coordinate
#define S_SP (255.0f / 511.0f)          // full-grid coordinate -> dgs index
#define TOL 1e-4f

typedef __attribute__((ext_vector_type(2))) float v2f;
typedef __attribute__((ext_vector_type(8))) float v8f;

// ---------------------------------------------------------------------------
// Helpers
// ---------------------------------------------------------------------------
// Zero-padded ('SAME' conv) load, branchless: clamp address, mask value.
__device__ __forceinline__ float ld_pad0(const float* __restrict__ p, int y, int x) {
    int yc = min(max(y, 0), DH - 1);
    int xc = min(max(x, 0), DW - 1);
    float v = p[yc * DW + xc];
    return ((y == yc) & (x == xc)) ? v : 0.0f;
}

// Unnormalized Gaussian band entry k[d] for d in [0,18], else 0.  Branchless.
__device__ __forceinline__ float gw(int d) {
    float t = (float)(d - RAD) * (1.0f / 3.0f);
    float w = __expf(-0.5f * t * t);
    return ((unsigned)d < 19u) ? w : 0.0f;
}

// 1 / sum of the 19 taps (static unrolled -> registers only).
__device__ __forceinline__ float gauss_inv_sum() {
    float s = 0.0f;
    #pragma unroll
    for (int t = 0; t < TAPS; ++t) {
        float d = (float)(t - RAD) * (1.0f / 3.0f);
        s += __expf(-0.5f * d * d);
    }
    return 1.0f / s;
}

// Per-lane constant band registers: chunk c holds k[4c+kh   - lm] and
// k[4c+kh+1 - lm], normalized.  Identical formula serves the A matrix of the
// vertical pass and the B matrix of the horizontal pass.
__device__ __forceinline__ void band_regs(v2f* kr, int kh, int lm) {
    float inv = gauss_inv_sum();
    #pragma unroll
    for (int c = 0; c < KCH; ++c) {
        int m0 = 4 * c + kh;
        kr[c].x = gw(m0 - lm) * inv;
        kr[c].y = gw(m0 + 1 - lm) * inv;
    }
}

// Plain bilinear on a DHxDW plane; (py,px) pre-clamped to [0, 255].
__device__ __forceinline__ float bilin256(const float* __restrict__ p, float py, float px) {
    float y0f = floorf(py), x0f = floorf(px);
    int y0 = (int)y0f, x0 = (int)x0f;
    int y1 = min(y0 + 1, DH - 1), x1 = min(x0 + 1, DW - 1);
    float fy = py - y0f, fx = px - x0f;
    float v00 = p[y0 * DW + x0], v01 = p[y0 * DW + x1];
    float v10 = p[y1 * DW + x0], v11 = p[y1 * DW + x1];
    return v00 * (1.f - fy) * (1.f - fx) + v01 * (1.f - fy) * fx
         + v10 * fy * (1.f - fx)         + v11 * fy * fx;
}

// ---------------------------------------------------------------------------
// Kernel 1: vertical Gaussian blur as WMMA band-GEMM.
// Per tile: out(16x16) = A(16x36 band, register-resident) x B(36x16 slab)
// A[i][m] = k[m-i]; B[m][j] = in[y0+m-9][x0+j] (zero padded).
// Each wave handles TPW consecutive tiles reusing the same A registers.
// ---------------------------------------------------------------------------
__global__ void dvae_blur_v(const float* __restrict__ in, float* __restrict__ out) {
    int wave = threadIdx.x >> 5;
    int lane = threadIdx.x & 31;
    int lm   = lane & 15;            // M (A row) / N (B col, D col)
    int kh   = (lane >> 4) << 1;     // 0 or 2: K sub-index pair this lane holds
    int mh   = (lane >> 4) << 3;     // +8 rows for upper half-wave (D layout)

    v2f aReg[KCH];
    band_regs(aReg, kh, lm);         // launch-wide constant A matrix

    int tile0 = (blockIdx.x * 8 + wave) * TPW;     // 32768 tiles total
    #pragma unroll
    for (int t = 0; t < TPW; ++t) {
        int tile  = tile0 + t;
        int plane = tile >> 8;                     // b*2+c
        int tip   = tile & 255;
        int y0 = (tip >> 4) << 4;
        int x0 = (tip & 15) << 4;

        const float* src = in  + (size_t)plane * PLANE;
        float*       dst = out + (size_t)plane * PLANE;

        v8f acc = {};
        #pragma unroll
        for (int c = 0; c < KCH; ++c) {
            int m0 = 4 * c + kh;
            v2f b;
            b.x = ld_pad0(src, y0 + m0     - RAD, x0 + lm);
            b.y = ld_pad0(src, y0 + m0 + 1 - RAD, x0 + lm);
            acc = __builtin_amdgcn_wmma_f32_16x16x4_f32(
                false, aReg[c], false, b, (short)0, acc, false, false);
        }

        #pragma unroll
        for (int r = 0; r < 8; ++r)
            dst[(y0 + r + mh) * DW + x0 + lm] = acc[r];
    }
}

// ---------------------------------------------------------------------------
// Kernel 2: horizontal Gaussian blur as WMMA band-GEMM.
// Per tile: out(16x16) = A(16x36 input slab) x B(36x16 band, registers)
// A[i][m] = in[y0+i][x0+m-9]; B[m][j] = k[m-j].
// ---------------------------------------------------------------------------
__global__ void dvae_blur_h(const float* __restrict__ in, float* __restrict__ out) {
    int wave = threadIdx.x >> 5;
    int lane = threadIdx.x & 31;
    int lm   = lane & 15;
    int kh   = (lane >> 4) << 1;
    int mh   = (lane >> 4) << 3;

    v2f bReg[KCH];
    band_regs(bReg, kh, lm);         // launch-wide constant B matrix

    int tile0 = (blockIdx.x * 8 + wave) * TPW;
    #pragma unroll
    for (int t = 0; t < TPW; ++t) {
        int tile  = tile0 + t;
        int plane = tile >> 8;
        int tip   = tile & 255;
        int y0 = (tip >> 4) << 4;
        int x0 = (tip & 15) << 4;

        const float* src = in  + (size_t)plane * PLANE;
        float*       dst = out + (size_t)plane * PLANE;

        v8f acc = {};
        #pragma unroll
        for (int c = 0; c < KCH; ++c) {
            int m0 = 4 * c + kh;
            v2f a;
            a.x = ld_pad0(src, y0 + lm, x0 + m0     - RAD);
            a.y = ld_pad0(src, y0 + lm, x0 + m0 + 1 - RAD);
            acc = __builtin_amdgcn_wmma_f32_16x16x4_f32(
                false, a, false, bReg[c], (short)0, acc, false, false);
        }

        #pragma unroll
        for (int r = 0; r < 8; ++r)
            dst[(y0 + r + mh) * DW + x0 + lm] = acc[r];
    }
}

// ---------------------------------------------------------------------------
// Kernel 3: x = grid + v / 2^NTP   (grid in full-grid units)
// ---------------------------------------------------------------------------
__global__ void dvae_init(const float* __restrict__ v, float* __restrict__ x) {
    int idx = blockIdx.x * blockDim.x + threadIdx.x;       // < FIELD_ELEMS
    int j  = idx & 255;
    int i  = (idx >> 8) & 255;
    int ch = (idx >> 16) & 1;
    float g = (float)(ch == 0 ? i : j) * G_SP;
    x[idx] = g + v[idx] * (1.0f / 256.0f);
}

// ---------------------------------------------------------------------------
// Kernel 4: one Euler step.  disp = x - grid; x += DT * bilinear(disp, x*s)
// One thread handles both channels of one (b,i,j) site.  L2-resident gathers.
// ---------------------------------------------------------------------------
__global__ void dvae_step(const float* __restrict__ xin, float* __restrict__ xout) {
    int idx = blockIdx.x * blockDim.x + threadIdx.x;       // < BATCH*PLANE
    int j = idx & 255;
    int i = (idx >> 8) & 255;
    int b = idx >> 16;

    const float* xb = xin + (size_t)b * NCH * PLANE;       // ch0 = y coords
    const float* xc = xb + PLANE;                          // ch1 = x coords
    int off = i * DW + j;
    float xy = xb[off];
    float xx = xc[off];

    float py = fminf(fmaxf(xy * S_SP, 0.0f), 255.0f);
    float px = fminf(fmaxf(xx * S_SP, 0.0f), 255.0f);
    float y0f = floorf(py), x0f = floorf(px);
    int y0 = (int)y0f, x0 = (int)x0f;
    int y1 = min(y0 + 1, DH - 1), x1 = min(x0 + 1, DW - 1);
    float fy = py - y0f, fx = px - x0f;
    float w00 = (1.f - fy) * (1.f - fx), w01 = (1.f - fy) * fx;
    float w10 = fy * (1.f - fx),         w11 = fy * fx;

    // displacement = x-field minus identity grid, evaluated at the 4 corners
    float dy = (xb[y0 * DW + x0] - y0 * G_SP) * w00 + (xb[y0 * DW + x1] - y0 * G_SP) * w01
             + (xb[y1 * DW + x0] - y1 * G_SP) * w10 + (xb[y1 * DW + x1] - y1 * G_SP) * w11;
    float dx = (xc[y0 * DW + x0] - x0 * G_SP) * w00 + (xc[y0 * DW + x1] - x1 * G_SP) * w01
             + (xc[y1 * DW + x0] - x0 * G_SP) * w10 + (xc[y1 * DW + x1] - x1 * G_SP) * w11;

    float* ob = xout + (size_t)b * NCH * PLANE;
    ob[off]         = xy + DT_C * dy;
    ob[PLANE + off] = xx + DT_C * dx;
}

// ---------------------------------------------------------------------------
// Kernel 5: bilinear upsample field to 512x512 (half-pixel) + warp atlas.
// Atlas values clamped to [TOL, 1-TOL] per corner (clip commutes with lerp).
// ---------------------------------------------------------------------------
__global__ void dvae_warp(const float* __restrict__ field,
                          const float* __restrict__ atlas,
                          float* __restrict__ out) {
    int idx = blockIdx.x * blockDim.x + threadIdx.x;       // < BATCH*GH*GW
    int X = idx & 511;
    int Y = (idx >> 9) & 511;
    int b = idx >> 18;

    // half-pixel-center bilinear resize 256 -> 512, edge clamp
    float py = fminf(fmaxf((float)Y * 0.5f - 0.25f, 0.0f), 255.0f);
    float px = fminf(fmaxf((float)X * 0.5f - 0.25f, 0.0f), 255.0f);

    const float* fb = field + (size_t)b * NCH * PLANE;
    float fyc = bilin256(fb,          py, px);             // full-grid y coord
    float fxc = bilin256(fb + PLANE,  py, px);             // full-grid x coord

    // sample clipped atlas at (fyc, fxc), border clamp
    float ay = fminf(fmaxf(fyc, 0.0f), 511.0f);
    float ax = fminf(fmaxf(fxc, 0.0f), 511.0f);
    float y0f = floorf(ay), x0f = floorf(ax);
    int y0 = (int)y0f, x0 = (int)x0f;
    int y1 = min(y0 + 1, GH - 1), x1 = min(x0 + 1, GW - 1);
    float fy = ay - y0f, fx = ax - x0f;

    float v00 = fminf(fmaxf(atlas[y0 * GW + x0], TOL), 1.0f - TOL);
    float v01 = fminf(fmaxf(atlas[y0 * GW + x1], TOL), 1.0f - TOL);
    float v10 = fminf(fmaxf(atlas[y1 * GW + x0], TOL), 1.0f - TOL);
    float v11 = fminf(fmaxf(atlas[y1 * GW + x1], TOL), 1.0f - TOL);

    out[idx] = v00 * (1.f - fy) * (1.f - fx) + v01 * (1.f - fy) * fx
             + v10 * fy * (1.f - fx)         + v11 * fy * fx;
}

// ---------------------------------------------------------------------------
// Launcher
// ---------------------------------------------------------------------------
extern "C" void kernel_launch(void* const* d_in, const int* in_sizes, int n_in,
                              void* d_out, int out_size, void* d_ws, size_t ws_size,
                              hipStream_t stream) {
    (void)in_sizes; (void)n_in; (void)out_size; (void)ws_size;
    const float* v_star = (const float*)d_in[0];   // [64,2,256,256]
    const float* atlas  = (const float*)d_in[1];   // [1,1,512,512]
    float* out  = (float*)d_out;                   // [64,1,512,512]
    float* bufA = (float*)d_ws;                    // FIELD_ELEMS floats
    float* bufB = bufA + FIELD_ELEMS;              // FIELD_ELEMS floats

    // 32768 16x16 tiles, 8 waves/block, TPW tiles per wave
    dvae_blur_v<<<32768 / (8 * TPW), 256, 0, stream>>>(v_star, bufA);
    dvae_blur_h<<<32768 / (8 * TPW), 256, 0, stream>>>(bufA, bufB);

    dvae_init<<<FIELD_ELEMS / 256, 256, 0, stream>>>(bufB, bufA);

    float* a = bufA;
    float* b = bufB;
    for (int s = 0; s < 8; ++s) {            // NTP Euler steps, ping-pong
        dvae_step<<<(BATCH * PLANE) / 256, 256, 0, stream>>>(a, b);
        float* t = a; a = b; b = t;
    }
    // 8 swaps -> result back in bufA (== a)

    dvae_warp<<<(BATCH * GH * GW) / 256, 256, 0, stream>>>(a, atlas, out);
}